// pyPD_25967372271856
// MI455X (gfx1250) — compile-verified
//
#include <hip/hip_runtime.h>

typedef __attribute__((ext_vector_type(2))) float v2f;
typedef __attribute__((ext_vector_type(8))) float v8f;

#define DT  (1.0f / 60.0f)
#define DT2 (DT * DT)
#define N_ITER 5
#define N_JACOBI 16   // Jacobi contraction ~0.25-0.4/sweep -> < fp32 eps after 16

#define ASYNC_WAIT(n) asm volatile("s_wait_asynccnt %0" :: "n"(n) : "memory")

// ---------------------------------------------------------------------------
// GEMM: Y (Nx16) = L (NxK, row-major) * X (Kx16), fp32 WMMA 16x16x4.
// One block per 16-row tile of L; 8 waves split K (512 each).
// Each wave double-buffers its X K-chunks (64 K-rows = 4KB, contiguous in
// global) into LDS via global_load_async_to_lds_b128 (ASYNCcnt), overlapping
// the copy with WMMA on the previous chunk. B-operand then comes from LDS.
// Partial 16x16 accumulators are summed through LDS (overlaid on the
// staging buffer after a barrier).
// ---------------------------------------------------------------------------
__global__ __launch_bounds__(256) void gemm_Lx16_k(
    const float* __restrict__ Lm, const float* __restrict__ X,
    float* __restrict__ Y, int K) {
  // 8 waves x 2 buffers x (64 K-rows * 16 cols) = 64KB
  __shared__ __align__(16) float xbuf[8][2][1024];

  const int tile  = blockIdx.x;            // 16-row tile index
  const int lane  = threadIdx.x & 31;
  const int wave  = threadIdx.x >> 5;
  const int row   = (tile << 4) + (lane & 15);
  const int nsel  = lane & 15;
  const int khalf = (lane >> 4) << 1;      // 0 or 2
  const int kchunk = K >> 3;               // 512 per wave for K=4096
  const int kw0    = wave * kchunk;        // this wave's K start
  const float* __restrict__ lrow = Lm + (size_t)row * (size_t)K;

  // ---- stage chunk 0 of this wave's X panel into LDS (async) ----
  {
    unsigned lds = (unsigned)(size_t)&xbuf[wave][0][0] + (unsigned)(lane * 16);
    unsigned long long ga =
        (unsigned long long)(size_t)(X + (size_t)kw0 * 16) + (unsigned long long)(lane * 16);
#pragma unroll
    for (int i = 0; i < 8; ++i)
      asm volatile("global_load_async_to_lds_b128 %0, %1, off offset:%2"
                   :: "v"(lds), "v"(ga), "n"(i * 512) : "memory");
  }

  v8f acc = {};
  for (int so = 0; so < 8; ++so) {
    const int cur = so & 1;
    if (so < 7) {
      // stage next chunk while computing on the current one
      unsigned lds = (unsigned)(size_t)&xbuf[wave][cur ^ 1][0] + (unsigned)(lane * 16);
      unsigned long long ga =
          (unsigned long long)(size_t)(X + (size_t)(kw0 + (so + 1) * 64) * 16) +
          (unsigned long long)(lane * 16);
#pragma unroll
      for (int i = 0; i < 8; ++i)
        asm volatile("global_load_async_to_lds_b128 %0, %1, off offset:%2"
                     :: "v"(lds), "v"(ga), "n"(i * 512) : "memory");
      ASYNC_WAIT(8);   // the 8 older loads (current chunk) have completed
    } else {
      ASYNC_WAIT(0);   // last chunk: drain
    }

    const float* bb = &xbuf[wave][cur][0];
    int kl = khalf;                 // local K within chunk
    int kg = kw0 + so * 64 + khalf; // global K for the A operand
    __builtin_prefetch(lrow + kg + 128, 0, 0);   // next 512B of this L row
#pragma unroll
    for (int si = 0; si < 16; ++si, kl += 4, kg += 4) {
      v2f a = *(const v2f*)(lrow + kg);   // A: K=kg, kg+1 (64-bit load)
      v2f b;
      b.x = bb[kl * 16 + nsel];           // B from LDS: K=kl
      b.y = bb[(kl + 1) * 16 + nsel];     // K=kl+1
      acc = __builtin_amdgcn_wmma_f32_16x16x4_f32(
          false, a, false, b, (short)0, acc, false, false);
    }
  }

  // ---- reduce 8 wave-partials; reuse the staging buffer as scratch ----
  float* red = &xbuf[0][0][0];     // needs 8KB of the 64KB buffer
  __syncthreads();                 // all waves done reading their xbuf regions
#pragma unroll
  for (int r = 0; r < 8; ++r) red[wave * 256 + r * 32 + lane] = acc[r];
  __syncthreads();

  const int t = threadIdx.x;       // 256 threads: each sums 8 wave-partials
  float s = 0.f;
#pragma unroll
  for (int wv = 0; wv < 8; ++wv) s += red[wv * 256 + t];
  const int r = t >> 5, l = t & 31;
  const int m = r + ((l >> 4) << 3);
  const int n = l & 15;
  Y[(size_t)((tile << 4) + m) * 16 + n] = s;
}

// ---------------------------------------------------------------------------
// Setup: y = x + dt*v (16-wide padded layout), xt = y,
//        m = diag(M), Adiag = m + dt^2 * diag(L).
// ---------------------------------------------------------------------------
__global__ void init_k(const float* __restrict__ x, const float* __restrict__ v,
                       const float* __restrict__ M, const float* __restrict__ Lm,
                       float* __restrict__ xt16, float* __restrict__ y16,
                       float* __restrict__ mdiag, float* __restrict__ Adiag,
                       int n) {
  int i = blockIdx.x * blockDim.x + threadIdx.x;
  if (i >= n) return;
  float mi = M[(size_t)i * n + i];
  mdiag[i] = mi;
  Adiag[i] = mi + DT2 * Lm[(size_t)i * n + i];
#pragma unroll
  for (int c = 0; c < 16; ++c) {
    float val = (c < 3) ? x[i * 3 + c] + DT * v[i * 3 + c] : 0.f;
    y16[i * 16 + c]  = val;
    xt16[i * 16 + c] = val;
  }
}

__global__ void zero_k(float* __restrict__ p, int n) {
  int i = blockIdx.x * blockDim.x + threadIdx.x;
  if (i < n) p[i] = 0.f;
}

// Per-edge spring projection + scatter-add of w*R into JtR (fp32 atomics).
__global__ void spring_k(const int* __restrict__ edges,
                         const float* __restrict__ w, const float* __restrict__ l0,
                         const float* __restrict__ xt16, float* __restrict__ JtR16,
                         int E) {
  int e = blockIdx.x * blockDim.x + threadIdx.x;
  if (e >= E) return;
  int j = edges[2 * e], k = edges[2 * e + 1];
  float d0 = xt16[j * 16 + 0] - xt16[k * 16 + 0];
  float d1 = xt16[j * 16 + 1] - xt16[k * 16 + 1];
  float d2 = xt16[j * 16 + 2] - xt16[k * 16 + 2];
  float nrm = sqrtf(d0 * d0 + d1 * d1 + d2 * d2);
  float s = w[e] * l0[e] / fmaxf(nrm, 1e-8f);
  float r0 = s * d0, r1 = s * d1, r2 = s * d2;
  atomicAdd(&JtR16[j * 16 + 0], r0);
  atomicAdd(&JtR16[j * 16 + 1], r1);
  atomicAdd(&JtR16[j * 16 + 2], r2);
  atomicAdd(&JtR16[k * 16 + 0], -r0);
  atomicAdd(&JtR16[k * 16 + 1], -r1);
  atomicAdd(&JtR16[k * 16 + 2], -r2);
}

// gf = m*(xt-y) + dt^2*(L@xt - JtR); first Jacobi sweep dx0 = gf / Adiag.
__global__ void gf_k(const float* __restrict__ xt16, const float* __restrict__ y16,
                     const float* __restrict__ Lx16, const float* __restrict__ JtR16,
                     const float* __restrict__ mdiag, const float* __restrict__ Adiag,
                     float* __restrict__ gf16, float* __restrict__ dx16, int n16) {
  int idx = blockIdx.x * blockDim.x + threadIdx.x;
  if (idx >= n16) return;
  int i = idx >> 4, c = idx & 15;
  float g = 0.f;
  if (c < 3)
    g = mdiag[i] * (xt16[idx] - y16[idx]) + DT2 * (Lx16[idx] - JtR16[idx]);
  gf16[idx] = g;
  dx16[idx] = g / Adiag[i];            // pad cols stay exactly 0
}

// Jacobi sweep: dx += (gf - (m*dx + dt^2 * L@dx)) / Adiag
__global__ void jacobi_k(const float* __restrict__ gf16, const float* __restrict__ Ldx16,
                         const float* __restrict__ mdiag, const float* __restrict__ Adiag,
                         float* __restrict__ dx16, int n16) {
  int idx = blockIdx.x * blockDim.x + threadIdx.x;
  if (idx >= n16) return;
  int i = idx >> 4;
  float dx = dx16[idx];
  float Adx = mdiag[i] * dx + DT2 * Ldx16[idx];
  dx16[idx] = dx + (gf16[idx] - Adx) / Adiag[i];
}

__global__ void upd_k(float* __restrict__ xt16, const float* __restrict__ dx16, int n16) {
  int idx = blockIdx.x * blockDim.x + threadIdx.x;
  if (idx < n16) xt16[idx] -= dx16[idx];
}

__global__ void out_k(const float* __restrict__ xt16, float* __restrict__ out, int n3) {
  int idx = blockIdx.x * blockDim.x + threadIdx.x;
  if (idx >= n3) return;
  int i = idx / 3, c = idx - 3 * i;
  out[idx] = xt16[i * 16 + c];
}

extern "C" void kernel_launch(void* const* d_in, const int* in_sizes, int n_in,
                              void* d_out, int out_size, void* d_ws, size_t ws_size,
                              hipStream_t stream) {
  (void)n_in; (void)out_size; (void)ws_size;
  const float* x     = (const float*)d_in[0];
  const float* v     = (const float*)d_in[1];
  const float* M     = (const float*)d_in[2];
  const float* Lm    = (const float*)d_in[3];
  const float* w     = (const float*)d_in[4];
  const float* l0    = (const float*)d_in[5];
  const int*   edges = (const int*)d_in[6];

  const int n   = in_sizes[0] / 3;   // 4096
  const int E   = in_sizes[4];       // 32768
  const int n16 = n * 16;
  const int n3  = n * 3;

  float* ws    = (float*)d_ws;
  float* xt16  = ws;                 // n*16
  float* y16   = xt16 + n16;         // n*16
  float* JtR16 = y16 + n16;          // n*16
  float* gf16  = JtR16 + n16;        // n*16
  float* dx16  = gf16 + n16;         // n*16
  float* Lx16  = dx16 + n16;         // n*16 (shared L@xt / L@dx buffer)
  float* mdiag = Lx16 + n16;         // n
  float* Adiag = mdiag + n;          // n

  const int B = 256;
  const int gN   = (n + B - 1) / B;
  const int g16  = (n16 + B - 1) / B;
  const int gE   = (E + B - 1) / B;
  const int g3   = (n3 + B - 1) / B;
  const int gTil = n / 16;           // 256 row-tiles for the WMMA GEMM

  init_k<<<gN, B, 0, stream>>>(x, v, M, Lm, xt16, y16, mdiag, Adiag, n);

  for (int it = 0; it < N_ITER; ++it) {
    zero_k<<<g16, B, 0, stream>>>(JtR16, n16);
    spring_k<<<gE, B, 0, stream>>>(edges, w, l0, xt16, JtR16, E);

    gemm_Lx16_k<<<gTil, B, 0, stream>>>(Lm, xt16, Lx16, n);   // L @ xt
    gf_k<<<g16, B, 0, stream>>>(xt16, y16, Lx16, JtR16, mdiag, Adiag, gf16, dx16, n16);

    for (int js = 1; js < N_JACOBI; ++js) {
      gemm_Lx16_k<<<gTil, B, 0, stream>>>(Lm, dx16, Lx16, n); // L @ dx
      jacobi_k<<<g16, B, 0, stream>>>(gf16, Lx16, mdiag, Adiag, dx16, n16);
    }
    upd_k<<<g16, B, 0, stream>>>(xt16, dx16, n16);
  }

  out_k<<<g3, B, 0, stream>>>(xt16, (float*)d_out, n3);
}